// TimeStretch_39273180955651
// MI455X (gfx1250) — compile-verified
//
#include <hip/hip_runtime.h>
#include <math.h>

// ---------------------------------------------------------------------------
// Phase vocoder time-stretch, MI455X (gfx1250, wave32).
//   B=16, F=1025, T=2048, rate=0.9 -> T_OUT=2276. Rows = B*F = 16400.
//   One wave per row; cumsum over time = lower-triangular-ones matmul done
//   with 4 chained V_WMMA_F32_16X16X4_F32 per 16x16 tile (exact f32).
//   Cross-column scan via v_permlane16 (VALU, no LDS round-trips).
//   Tiles 0..7 are branch-free (bounds proven); only the tail tile clamps.
// ---------------------------------------------------------------------------

typedef float v2f __attribute__((ext_vector_type(2)));
typedef float v8f __attribute__((ext_vector_type(8)));

#define NB      16
#define NF      1025
#define T_IN    2048
#define T_OUT   2276          // ceil(2048 / 0.9)
#define RATE    0.9f
#define ROWS    (NB * NF)     // 16400
#define WPB     8             // waves per 256-thread block
#define TILE    256           // output frames per wave-iteration (16x16)
#define NTILES  9             // ceil(2276 / 256)

// v_permlane16_b32 with compile-time selects (per-16-lane-row permute).
__device__ __forceinline__ float plane16(float x, unsigned s0, unsigned s1) {
    return __int_as_float(__builtin_amdgcn_permlane16(
        __float_as_int(x), __float_as_int(x), s0, s1, false, false));
}
// v_permlanex16_b32 identity: each lane reads the opposite 16-lane row, same slot.
__device__ __forceinline__ float planex16_id(float x) {
    return __int_as_float(__builtin_amdgcn_permlanex16(
        __float_as_int(x), __float_as_int(x), 0x76543210u, 0xFEDCBA98u, false, false));
}

// Fast atan2 (finite inputs; (0,0)->0). Cephes atanf reduction + polynomial,
// max error ~1.2e-7 rad. Uses raw v_rcp_f32 (<=1 ulp) twice.
__device__ __forceinline__ float fast_atan2f(float y, float x) {
    const float ax = fabsf(x), ay = fabsf(y);
    const float mx = fmaxf(ax, ay);
    const float mn = fminf(ax, ay);
    float a = mn * __builtin_amdgcn_rcpf(mx);       // tan of angle in [0,1]
    a = (mn == 0.0f) ? 0.0f : a;                    // handles (0,0)
    // second reduction: a > tan(pi/8) -> z = (a-1)/(a+1), add pi/4
    const bool  red = (a > 0.4142135624f);
    const float t   = (a - 1.0f) * __builtin_amdgcn_rcpf(a + 1.0f);
    const float z   = red ? t : a;                  // |z| <= 0.4142
    const float s   = z * z;
    float p = fmaf(s, 8.05374449538e-2f, -1.38776856032e-1f);
    p = fmaf(s, p, 1.99777106478e-1f);
    p = fmaf(s, p, -3.33329491539e-1f);
    float r = fmaf(z * s, p, z);
    r = red ? (r + 0.7853981634f) : r;              // atan(mn/mx) in [0, pi/4]
    r = (ay > ax)  ? (1.5707963268f - r) : r;       // true octant
    r = (x < 0.0f) ? (3.1415926536f - r) : r;       // left half-plane
    return copysignf(r, y);
}

// delta[t] for t>=1: wrap(angle(s1[t-1]) - angle(s0[t-1])) == angle(s1*conj(s0)).
// Bounds (no guards needed for !TAIL): tiles 0..7 -> u<=2046 -> i0+1<=1842.
// TAIL: clamp u so i0+1<=2047; zero the result for t>=T_OUT.
// t==0 (u=-1): (int)(-0.9f)==0, load-safe; caller patches with delta0.
template <bool TAIL>
__device__ __forceinline__ float pv_delta(int t, const float* __restrict__ rr,
                                          const float* __restrict__ ri) {
    int u = t - 1;
    if (TAIL) u = (u > T_OUT - 2) ? (T_OUT - 2) : u;
    const float    ts = (float)u * RATE;
    const unsigned i0 = (unsigned)(int)ts;          // >=0: zext -> saddr form
    const float r0 = rr[i0],     q0 = ri[i0];
    const float r1 = rr[i0 + 1], q1 = ri[i0 + 1];
    const float dot = fmaf(r1, r0, q1 * q0);
    const float crs = fmaf(q1, r0, -(r1 * q0));
    float d = fast_atan2f(crs, dot);
    if (TAIL) d = (t < T_OUT) ? d : 0.0f;
    return d;
}

// One 16x16 output tile for this wave's row. Returns updated phase carry.
template <bool TAIL>
__device__ __forceinline__ float pv_tile(int base, float carry, float d0,
                                         int lane, int nn, bool hi,
                                         const float* __restrict__ rr,
                                         const float* __restrict__ ri,
                                         float* __restrict__ po) {
    // ---- intra-column inclusive scan: D = Ltri(16x16) x Delta(16x16) ------
    v8f acc = {0.f, 0.f, 0.f, 0.f, 0.f, 0.f, 0.f, 0.f};
#pragma unroll
    for (int c = 0; c < 4; ++c) {
        const int k0 = 4 * c + (hi ? 2 : 0);
        v2f a, b;
        // A (16x4 tri slice): lane -> M = lane&15 ; v0/v1 -> K = k0 / k0+1
        a.x = (k0     <= nn) ? 1.0f : 0.0f;
        a.y = (k0 + 1 <= nn) ? 1.0f : 0.0f;
        // B (4x16 deltas): lane -> N = nn ; v0/v1 -> K rows k0 / k0+1
        b.x = pv_delta<TAIL>(base + nn * 16 + k0,     rr, ri);
        b.y = pv_delta<TAIL>(base + nn * 16 + k0 + 1, rr, ri);
        if (!TAIL && c == 0)                 // t==0 slot: initial phase fixup
            b.x = (base == 0 && lane == 0) ? d0 : b.x;
        acc = __builtin_amdgcn_wmma_f32_16x16x4_f32(
            false, a, false, b, (short)0, acc, false, false);
    }

    // ---- cross-column exclusive scan of the 16 column sums (pure VALU) ----
    // acc[7] on hi lanes = D[15][nn] = sum of column nn.
    const float own = acc[7];
    const float oth = planex16_id(own);          // opposite row, same nn
    const float cs  = hi ? own : oth;            // colsum[nn] in every lane
    float incl = cs;
    { const float o = plane16(incl, 0x6543210Fu, 0xEDCBA987u);  // shift 1
      incl += (nn >= 1) ? o : 0.0f; }
    { const float o = plane16(incl, 0x543210FEu, 0xDCBA9876u);  // shift 2
      incl += (nn >= 2) ? o : 0.0f; }
    { const float o = plane16(incl, 0x3210FEDCu, 0xBA987654u);  // shift 4
      incl += (nn >= 4) ? o : 0.0f; }
    { const float o = plane16(incl, 0xFEDCBA98u, 0x76543210u);  // shift 8
      incl += (nn >= 8) ? o : 0.0f; }
    const float off = carry + (incl - cs);       // phase offset for column nn
    carry += plane16(incl, 0xFFFFFFFFu, 0xFFFFFFFFu);  // broadcast row-lane 15

    // ---- magnitudes + complex synthesis (C/D layout), stores --------------
    // lane -> N = nn ; VGPR j -> M = j (lo half) / j+8 (hi half)
#pragma unroll
    for (int j = 0; j < 8; ++j) {
        const int m = j + (hi ? 8 : 0);
        const int t = base + nn * 16 + m;
        const int tc = TAIL ? ((t > T_OUT - 1) ? (T_OUT - 1) : t) : t;
        const float    ts = (float)tc * RATE;
        const unsigned i0 = (unsigned)(int)ts;   // >=0: zext -> saddr form
        const float    al = ts - (float)(int)i0;
        const float r0 = rr[i0], q0 = ri[i0];
        float r1, q1;
        if (TAIL) {
            const unsigned i1  = i0 + 1;
            const unsigned i1c = (i1 > T_IN - 1u) ? (T_IN - 1u) : i1;
            const float r1l = rr[i1c], q1l = ri[i1c];   // unconditional load
            const bool inb = (i1 < (unsigned)T_IN);
            r1 = inb ? r1l : 0.0f;                      // cndmask, no branch
            q1 = inb ? q1l : 0.0f;
        } else {
            r1 = rr[i0 + 1];
            q1 = ri[i0 + 1];
        }
        const float n0 = __builtin_amdgcn_sqrtf(fmaf(r0, r0, q0 * q0));
        const float n1 = __builtin_amdgcn_sqrtf(fmaf(r1, r1, q1 * q1));
        const float mg = fmaf(al, n1, (1.0f - al) * n0);
        const float ph = acc[j] + off;
        float s, c;
        __sincosf(ph, &s, &c);
        if (!TAIL || t < T_OUT) {
            po[2 * (unsigned)t]     = mg * c;
            po[2 * (unsigned)t + 1] = mg * s;
        }
    }
    return carry;
}

__global__ __launch_bounds__(256)
void pv_timestretch_kernel(const float* __restrict__ xr,
                           const float* __restrict__ xi,
                           float* __restrict__ out) {
    const int lane = threadIdx.x & 31;
    const int wave = threadIdx.x >> 5;
    // Wave-uniform row -> scalar (SADDR) addressing for all loads/stores.
    const int row  = __builtin_amdgcn_readfirstlane(blockIdx.x * WPB + wave);
    if (row >= ROWS) return;

    const float* rr = xr + (size_t)row * T_IN;
    const float* ri = xi + (size_t)row * T_IN;
    float*       po = out + (size_t)row * (size_t)(2 * T_OUT);

    const int  nn = lane & 15;                   // column N this lane owns
    const bool hi = (lane >= 16);

    const float d0 = fast_atan2f(ri[0], rr[0]);  // delta[0] = angle(spec[...,0])

    float carry = 0.0f;
#pragma unroll 1
    for (int tile = 0; tile < NTILES - 1; ++tile)
        carry = pv_tile<false>(tile * TILE, carry, d0, lane, nn, hi, rr, ri, po);
    pv_tile<true>((NTILES - 1) * TILE, carry, d0, lane, nn, hi, rr, ri, po);
}

extern "C" void kernel_launch(void* const* d_in, const int* in_sizes, int n_in,
                              void* d_out, int out_size, void* d_ws, size_t ws_size,
                              hipStream_t stream) {
    (void)in_sizes; (void)n_in; (void)d_ws; (void)ws_size; (void)out_size;
    const float* xr  = (const float*)d_in[0];
    const float* xi  = (const float*)d_in[1];
    float*       out = (float*)d_out;
    const int blocks = (ROWS + WPB - 1) / WPB;   // 2050
    pv_timestretch_kernel<<<blocks, 256, 0, stream>>>(xr, xi, out);
}